// GraphConvLayer_52518860095779
// MI455X (gfx1250) — compile-verified
//
#include <hip/hip_runtime.h>
#include <math.h>

#define EPSN 1e-12f
#define Bc   4
#define Nc   4096
#define Mc   32
#define Fc   64
#define NMc  (Nc*Mc)

typedef __attribute__((ext_vector_type(2))) float v2f;
typedef __attribute__((ext_vector_type(8))) float v8f;

static __device__ inline v8f wmma4(v2f a, v2f b, v8f c) {
  // D(16x16,f32) = A(16x4,f32) x B(4x16,f32) + C
  return __builtin_amdgcn_wmma_f32_16x16x4_f32(false, a, false, b, (short)0, c,
                                               false, false);
}

// Pair-interleaved weight layout: element (k, col) of a Kx64 weight goes to
// sW[(k>>1)*128 + col*2 + (k&1)], so the B fragment {W[k][col], W[k+1][col]}
// (k even) is a contiguous 8-byte-aligned float2 -> single ds_load_b64.
static __device__ inline int wswz(int k, int col) {
  return (k >> 1) * 128 + col * 2 + (k & 1);
}
static __device__ inline v2f bfrag(const float* sW, int k, int col) {
  return *(const v2f*)&sW[(k >> 1) * 128 + col * 2];
}

__global__ void k_zero(float* __restrict__ s, int n) {
  int i = blockIdx.x * blockDim.x + threadIdx.x;
  if (i < n) s[i] = 0.f;
}

// ---------------- Kernel A: atom_update = relu((root*nbr_weighted) @ Wn + bn)
__global__ __launch_bounds__(128) void k_atom(
    const float* __restrict__ atom, const float* __restrict__ bond,
    const int* __restrict__ adj, const float* __restrict__ Wnode,
    const float* __restrict__ bnode, float* __restrict__ atomU)
{
  __shared__ float sW[64 * 64];        // pair-interleaved
  __shared__ float sX[4][16 * 64];
  __shared__ float swt[4][16][32];
  __shared__ float sScale[4][16];

  const int tid = threadIdx.x;
  const int lane = tid & 31;
  const int wv = tid >> 5;
  const int b = blockIdx.y;
  const int n0 = (blockIdx.x * 4 + wv) * 16;

  for (int i = tid; i < 64 * 64; i += 128)
    sW[wswz(i >> 6, i & 63)] = Wnode[i];

  // per-(n,m) edge weights: (||bond||_2)^-2 = 1/sum(bond^2)
  const float* bondB = bond + (size_t)b * NMc * Fc;
  for (int i = 0; i < 16; ++i) {
    const float* p = bondB + ((size_t)(n0 + i) * Mc + lane) * Fc;
    float ss = 0.f;
#pragma unroll
    for (int f = 0; f < Fc; f += 4) {
      float4 v = *(const float4*)(p + f);
      ss += v.x * v.x + v.y * v.y + v.z * v.z + v.w * v.w;
    }
    swt[wv][i][lane] = 1.0f / ss;
  }
  __syncthreads();
  if (lane < 16) {
    float s = 0.f;
#pragma unroll
    for (int m = 0; m < Mc; ++m) s += swt[wv][lane][m];
    sScale[wv][lane] = 1.0f / fmaxf(s, EPSN);
  }
  __syncthreads();

  // X[n][f] = sqrt|atom[n,f]| * sum_m w * sqrt|atom[adj[n,m],f]|
  const float* atomB = atom + (size_t)b * Nc * Fc;
  const int* adjB = adj + (size_t)b * NMc;
  const int f0 = lane * 2;
  for (int n = 0; n < 16; ++n) {
    float a0 = 0.f, a1 = 0.f;
    const float rs = sScale[wv][n];
    for (int m = 0; m < Mc; ++m) {
      const int idx = adjB[(n0 + n) * Mc + m];
      const float w = swt[wv][n][m] * rs;
      const float* pr = atomB + (size_t)idx * Fc + f0;
      a0 += w * sqrtf(fabsf(pr[0]));
      a1 += w * sqrtf(fabsf(pr[1]));
    }
    const float* pc = atomB + (size_t)(n0 + n) * Fc + f0;
    sX[wv][n * 64 + f0]     = sqrtf(fabsf(pc[0])) * a0;
    sX[wv][n * 64 + f0 + 1] = sqrtf(fabsf(pc[1])) * a1;
  }
  __syncthreads();

  // 16x64 @ 64x64 via v_wmma_f32_16x16x4_f32
  const int r = lane & 15;
  const int koff = (lane < 16) ? 0 : 2;
  v8f acc[4] = {};
  for (int s = 0; s < 16; ++s) {
    const int k = 4 * s + koff;
    v2f a = *(const v2f*)&sX[wv][r * 64 + k];
#pragma unroll
    for (int nt = 0; nt < 4; ++nt)
      acc[nt] = wmma4(a, bfrag(sW, k, nt * 16 + r), acc[nt]);
  }
  const int hi = (lane < 16) ? 0 : 8;
  float* outB = atomU + (size_t)b * Nc * Fc;
#pragma unroll
  for (int nt = 0; nt < 4; ++nt) {
    const int col = nt * 16 + r;
    const float bias = bnode[col];
#pragma unroll
    for (int rr = 0; rr < 8; ++rr) {
      const int m = rr + hi;
      outB[(size_t)(n0 + m) * Fc + col] = fmaxf(acc[nt][rr] + bias, 0.f);
    }
  }
}

// ---------------- Kernel B: S[b][f] = sum_e |atomU[b, tup[e, f/64], f%64]|
__global__ __launch_bounds__(256) void k_colsum(
    const float* __restrict__ atomU, const int* __restrict__ tup,
    float* __restrict__ S)
{
  __shared__ float part[256];
  const int b = blockIdx.y;
  const int tid = threadIdx.x;
  const int f = tid & 127;
  const int half = tid >> 7;
  const int j = f >> 6;
  const int c = f & 63;
  const int eper = NMc / gridDim.x;
  const int e0 = blockIdx.x * eper;
  const float* aB = atomU + (size_t)b * Nc * Fc;
  const int* tB = tup + (size_t)b * NMc * 2;
  float s = 0.f;
  for (int e = e0 + half; e < e0 + eper; e += 2) {
    const int idx = tB[e * 2 + j];
    s += fabsf(aB[(size_t)idx * Fc + c]);
  }
  part[tid] = s;
  __syncthreads();
  if (tid < 128) atomicAdd(&S[b * 128 + f], part[tid] + part[tid + 128]);
}

// ---------------- Kernel C: fused edge pipeline
// bond_update = (bond + tanh( (gather/S) @ Wn2e + bn2e )) @ We + be
// L1-norm over edges folded into Wn2e rows: W'[k] = Wn2e[k] / max(S[k], eps)
__global__ __launch_bounds__(128) void k_bond(
    const float* __restrict__ atomU, const float* __restrict__ bond,
    const int* __restrict__ tup, const float* __restrict__ Wn2e,
    const float* __restrict__ We, const float* __restrict__ bn2e,
    const float* __restrict__ be, const float* __restrict__ S,
    float* __restrict__ out)
{
  __shared__ float sW1[128 * 64];   // 32 KB, prescaled by 1/S, pair-interleaved
  __shared__ float sW2[64 * 64];    // 16 KB, pair-interleaved
  __shared__ float sZ[4][16 * 64];  // 16 KB per-wave restripe tiles

  const int tid = threadIdx.x, lane = tid & 31, wv = tid >> 5;
  const int b = blockIdx.y;

  for (int i = tid; i < 128 * 64; i += 128) {
    const int k = i >> 6;
    const float inv = 1.0f / fmaxf(S[b * 128 + k], EPSN);
    sW1[wswz(k, i & 63)] = Wn2e[i] * inv;
  }
  for (int i = tid; i < 64 * 64; i += 128)
    sW2[wswz(i >> 6, i & 63)] = We[i];
  __syncthreads();

  const int t = blockIdx.x * 4 + wv;
  const int e0 = t * 16;
  const int r = lane & 15;
  const int koff = (lane < 16) ? 0 : 2;
  const int hi = (lane < 16) ? 0 : 8;

  const int* tB = tup + (size_t)b * NMc * 2;
  const int i0 = tB[(e0 + r) * 2 + 0];
  const int i1 = tB[(e0 + r) * 2 + 1];
  const float* aB = atomU + (size_t)b * Nc * Fc;
  const float* row0 = aB + (size_t)i0 * Fc;
  const float* row1 = aB + (size_t)i1 * Fc;

  // GEMM1: 16x128 @ 128x64 (32 k-steps of v_wmma_f32_16x16x4_f32)
  v8f acc[4] = {};
  for (int s = 0; s < 32; ++s) {
    const float* rp = (s < 16) ? row0 : row1;
    v2f a = *(const v2f*)&rp[4 * (s & 15) + koff];
    const int k = 4 * s + koff;
#pragma unroll
    for (int nt = 0; nt < 4; ++nt)
      acc[nt] = wmma4(a, bfrag(sW1, k, nt * 16 + r), acc[nt]);
  }

  // Z = bond + tanh(Y + bias); restripe through LDS into A-fragment order
  const float* bB = bond + (size_t)b * NMc * Fc;
#pragma unroll
  for (int nt = 0; nt < 4; ++nt) {
    const int col = nt * 16 + r;
    const float bias = bn2e[col];
#pragma unroll
    for (int rr = 0; rr < 8; ++rr) {
      const int m = rr + hi;
      const float y = tanhf(acc[nt][rr] + bias);
      sZ[wv][m * 64 + col] = y + bB[(size_t)(e0 + m) * Fc + col];
    }
  }
  __syncthreads();

  // GEMM2: 16x64 @ 64x64
  v8f acc2[4] = {};
  for (int s = 0; s < 16; ++s) {
    const int k = 4 * s + koff;
    v2f a = *(const v2f*)&sZ[wv][r * 64 + k];
#pragma unroll
    for (int nt = 0; nt < 4; ++nt)
      acc2[nt] = wmma4(a, bfrag(sW2, k, nt * 16 + r), acc2[nt]);
  }
  float* oB = out + (size_t)b * NMc * Fc;
#pragma unroll
  for (int nt = 0; nt < 4; ++nt) {
    const int col = nt * 16 + r;
    const float bias = be[col];
#pragma unroll
    for (int rr = 0; rr < 8; ++rr) {
      const int m = rr + hi;
      oB[(size_t)(e0 + m) * Fc + col] = acc2[nt][rr] + bias;
    }
  }
}

extern "C" void kernel_launch(void* const* d_in, const int* in_sizes, int n_in,
                              void* d_out, int out_size, void* d_ws, size_t ws_size,
                              hipStream_t stream)
{
  (void)in_sizes; (void)n_in; (void)out_size; (void)ws_size;
  const float* atom  = (const float*)d_in[0];
  const float* bond  = (const float*)d_in[1];
  const int*   adj   = (const int*)d_in[2];
  const int*   tup   = (const int*)d_in[3];
  const float* Wn    = (const float*)d_in[4];
  const float* We    = (const float*)d_in[5];
  const float* Wn2e  = (const float*)d_in[6];
  const float* bn    = (const float*)d_in[7];
  const float* bedge = (const float*)d_in[8];
  const float* bn2e  = (const float*)d_in[9];

  float* atomU = (float*)d_out;                            // (B,N,64)
  float* bondU = (float*)d_out + (size_t)Bc * Nc * Fc;     // (B,N,M,64)
  float* S     = (float*)d_ws;                             // (B,128) col sums

  hipLaunchKernelGGL(k_zero,   dim3(1),        dim3(512), 0, stream, S, Bc * 128);
  hipLaunchKernelGGL(k_atom,   dim3(64, Bc),   dim3(128), 0, stream,
                     atom, bond, adj, Wn, bn, atomU);
  hipLaunchKernelGGL(k_colsum, dim3(256, Bc),  dim3(256), 0, stream,
                     atomU, tup, S);
  hipLaunchKernelGGL(k_bond,   dim3(2048, Bc), dim3(128), 0, stream,
                     atomU, bond, tup, Wn2e, We, bn2e, bedge, S, bondU);
}